// Edge_to_Node_Conv2d_88167088653043
// MI455X (gfx1250) — compile-verified
//
#include <hip/hip_runtime.h>

typedef __attribute__((ext_vector_type(2))) float v2f;
typedef __attribute__((ext_vector_type(8))) float v8f;

#define K_DIM 1024
#define C_DIM 16
#define B_DIM 64

// ---------------------------------------------------------------------------
// Kernel 1: CcolT[b][i][c] = sum_j X[b][i][j] * W_col[c][j]  (+ b_col[c])
// WMMA mapping: D(16x16: M=i rows, N=c) = A(16x4: X rows) x B(4x16: W_col^T)
// One wave per (batch, 16-row strip). K-loop over j, 16 per iteration,
// software-pipelined: prefetch tile j+16 while WMMAs consume tile j.
// ---------------------------------------------------------------------------
__global__ __launch_bounds__(128) void ccol_kernel(const float* __restrict__ x,
                                                   const float* __restrict__ Wcol,
                                                   const float* __restrict__ bcol,
                                                   float* __restrict__ ccolT) {
    const int wave = (blockIdx.x << 2) + (threadIdx.x >> 5);   // 0..4095
    const int b    = wave >> 6;                                // batch
    const int i0   = (wave & 63) << 4;                         // row strip base
    const int lane = threadIdx.x & 31;
    const int half = lane >> 4;                                // 0 or 1
    const int mn   = lane & 15;                                // M for A, N for B
    const int kb   = half << 1;                                // K sub-offset {0,2}

    // A source: 16 rows of X (this lane's row), 2 consecutive f32 per step
    const float* xrow = x + (((size_t)b) << 20) + (size_t)(i0 + mn) * K_DIM + kb;
    // B source: W_col^T -> row c of W_col (this lane's column n = c)
    const float* wrow = Wcol + (size_t)mn * K_DIM + kb;

    v8f acc[4] = {};
    v2f A[4], Bm[4];

    // prologue: load tile j = 0
#pragma unroll
    for (int u = 0; u < 4; ++u) {
        A[u]  = *(const v2f*)(xrow + (u << 2));
        Bm[u] = *(const v2f*)(wrow + (u << 2));
    }

    // steady state: prefetch tile j, compute tile j-16
    for (int j = 16; j < K_DIM; j += 16) {
        v2f An[4], Bn[4];
#pragma unroll
        for (int u = 0; u < 4; ++u) {
            An[u] = *(const v2f*)(xrow + j + (u << 2));
            Bn[u] = *(const v2f*)(wrow + j + (u << 2));
        }
#pragma unroll
        for (int u = 0; u < 4; ++u)
            acc[u] = __builtin_amdgcn_wmma_f32_16x16x4_f32(
                false, A[u], false, Bm[u], (short)0, acc[u], false, false);
#pragma unroll
        for (int u = 0; u < 4; ++u) { A[u] = An[u]; Bm[u] = Bn[u]; }
    }

    // epilogue: compute final tile
#pragma unroll
    for (int u = 0; u < 4; ++u)
        acc[u] = __builtin_amdgcn_wmma_f32_16x16x4_f32(
            false, A[u], false, Bm[u], (short)0, acc[u], false, false);

    v8f d = (acc[0] + acc[1]) + (acc[2] + acc[3]);

    const float bc = bcol[mn];                                 // n = c
#pragma unroll
    for (int v = 0; v < 8; ++v) {
        const int i = i0 + v + (half << 3);                    // D-layout: M = v (+8)
        ccolT[(size_t)((b << 10) + i) * C_DIM + mn] = d[v] + bc;
    }
}

// ---------------------------------------------------------------------------
// Kernel 2: R[b][c][j] = sum_i W_row[c][i] * X[b][i][j]  (+ b_row[c])
// WMMA mapping: D(16x16: M=c, N=j) = A(16x4: W_row chunk) x B(4x16: X tile)
// One wave per (batch, 16-column tile). Same software pipelining over i.
// ---------------------------------------------------------------------------
__global__ __launch_bounds__(128) void row_kernel(const float* __restrict__ x,
                                                  const float* __restrict__ Wrow,
                                                  const float* __restrict__ brow,
                                                  float* __restrict__ r) {
    const int wave = (blockIdx.x << 2) + (threadIdx.x >> 5);   // 0..4095
    const int b    = wave >> 6;
    const int j0   = (wave & 63) << 4;                         // column tile base
    const int lane = threadIdx.x & 31;
    const int half = lane >> 4;
    const int mn   = lane & 15;                                // M=c for A, N=j for B
    const int kb   = half << 1;

    // A source: row c of W_row
    const float* arow = Wrow + (size_t)mn * K_DIM + kb;
    // B source: column j0+n of X (row stride K_DIM), starting at row kb
    const float* xcol = x + (((size_t)b) << 20) + ((size_t)kb << 10) + j0 + mn;

    v8f acc[4] = {};
    v2f A[4], Bm[4];

    // prologue: load tile i = 0
#pragma unroll
    for (int u = 0; u < 4; ++u) {
        A[u]   = *(const v2f*)(arow + (u << 2));
        Bm[u].x = xcol[(size_t)(u << 2) << 10];
        Bm[u].y = xcol[(size_t)((u << 2) + 1) << 10];
    }

    // steady state: prefetch tile i, compute tile i-16
    for (int i = 16; i < K_DIM; i += 16) {
        v2f An[4], Bn[4];
#pragma unroll
        for (int u = 0; u < 4; ++u) {
            const int ii = i + (u << 2);
            An[u]  = *(const v2f*)(arow + ii);
            Bn[u].x = xcol[(size_t)ii << 10];
            Bn[u].y = xcol[(size_t)(ii + 1) << 10];
        }
#pragma unroll
        for (int u = 0; u < 4; ++u)
            acc[u] = __builtin_amdgcn_wmma_f32_16x16x4_f32(
                false, A[u], false, Bm[u], (short)0, acc[u], false, false);
#pragma unroll
        for (int u = 0; u < 4; ++u) { A[u] = An[u]; Bm[u] = Bn[u]; }
    }

    // epilogue: compute final tile
#pragma unroll
    for (int u = 0; u < 4; ++u)
        acc[u] = __builtin_amdgcn_wmma_f32_16x16x4_f32(
            false, A[u], false, Bm[u], (short)0, acc[u], false, false);

    v8f d = (acc[0] + acc[1]) + (acc[2] + acc[3]);

#pragma unroll
    for (int v = 0; v < 8; ++v) {
        const int c = v + (half << 3);                         // D-layout: M = c
        r[((size_t)(b * C_DIM + c) << 10) + j0 + mn] = d[v] + brow[c];
    }
}

// ---------------------------------------------------------------------------
// Kernel 3: s[b][c] = sum_k R[b][c][k] * CcolT[b][k][c]; broadcast to (B,C,K)
// One block per batch, 512 threads = 16 waves; wave w owns channel c = w.
// ---------------------------------------------------------------------------
__global__ __launch_bounds__(512) void dot_kernel(const float* __restrict__ r,
                                                  const float* __restrict__ ccolT,
                                                  float* __restrict__ out) {
    const int b    = blockIdx.x;
    const int tid  = threadIdx.x;
    const int c    = tid >> 5;                                 // one channel per wave
    const int lane = tid & 31;

    const float* rp = r     + ((size_t)(b * C_DIM + c) << 10);
    const float* cp = ccolT + ((size_t)(b << 10)) * C_DIM + c;

    float partial = 0.0f;
    for (int k = lane; k < K_DIM; k += 32)
        partial += rp[k] * cp[(size_t)k * C_DIM];

#pragma unroll
    for (int off = 16; off > 0; off >>= 1)
        partial += __shfl_down(partial, off, 32);

    __shared__ float sv[C_DIM];
    if (lane == 0) sv[c] = partial;
    __syncthreads();

    float* ob = out + ((size_t)b * C_DIM << 10);
    for (int e = tid; e < C_DIM * K_DIM; e += 512)
        ob[e] = sv[e >> 10];
}

// ---------------------------------------------------------------------------
extern "C" void kernel_launch(void* const* d_in, const int* in_sizes, int n_in,
                              void* d_out, int out_size, void* d_ws, size_t ws_size,
                              hipStream_t stream) {
    const float* x     = (const float*)d_in[0];   // (64,1,1024,1024)
    const float* W_row = (const float*)d_in[1];   // (16,1024)
    const float* b_row = (const float*)d_in[2];   // (16,)
    const float* W_col = (const float*)d_in[3];   // (16,1024)
    const float* b_col = (const float*)d_in[4];   // (16,)
    float*       out   = (float*)d_out;           // (64,16,1024)

    float* Rws = (float*)d_ws;                          // 64*16*1024 f32 = 4 MB
    float* Cws = Rws + (size_t)B_DIM * C_DIM * K_DIM;   // 64*1024*16 f32 = 4 MB

    // 4096 waves each for the two GEMMs, 4 waves (128 threads) per block
    row_kernel <<<1024, 128, 0, stream>>>(x, W_row, b_row, Rws);
    ccol_kernel<<<1024, 128, 0, stream>>>(x, W_col, b_col, Cws);
    dot_kernel <<<B_DIM, 512, 0, stream>>>(Rws, Cws, out);
}